// DenseSigmoidInnerTransform_12292196401716
// MI455X (gfx1250) — compile-verified
//
#include <hip/hip_runtime.h>

typedef __attribute__((ext_vector_type(2))) float v2f;
typedef __attribute__((ext_vector_type(8))) float v8f;

#define BATCH_N  32768
#define NPARAM   544
#define WAVES_PB 8
// DEFAULT_U_A = log(e - 1 - 0.001)
#define DEF_UA   0.54074270f
#define LOG2E    1.44269504f
#define LN2      0.69314718f

// xor-lane exchange via DS_SWIZZLE_B32 (group-of-32: xor_mask<<10 | and_mask 0x1F)
template<int MASK>
__device__ __forceinline__ float swz_xor(float v) {
  return __int_as_float(
      __builtin_amdgcn_ds_swizzle(__float_as_int(v), (MASK << 10) | 0x1F));
}
__device__ __forceinline__ float fexp2(float v) { return __builtin_amdgcn_exp2f(v); }
__device__ __forceinline__ float flog2(float v) { return __builtin_amdgcn_logf(v); }
__device__ __forceinline__ float frcp(float v)  { return __builtin_amdgcn_rcpf(v); }

__global__ __launch_bounds__(256, 2)
void fused_dsit_kernel(const float* __restrict__ x,
                       const float* __restrict__ h,
                       float* __restrict__ out) {
  // per-wave LDS: U[16][17] | W'[16][17] | c[16] | g[16]  = 576 floats
  __shared__ float lds[WAVES_PB][576];
  const int lane = threadIdx.x & 31;
  const int wave = threadIdx.x >> 5;
  const int b    = blockIdx.x * WAVES_PB + wave;

  float* Um = &lds[wave][0];
  float* Wm = &lds[wave][272];
  float* cv = &lds[wave][544];
  float* gv = &lds[wave][560];

  const int r  = lane & 15;   // row this lane owns
  const int hf = lane >> 4;   // column half: 0 -> cols 0..7, 1 -> cols 8..15
  const float* hb = h + (size_t)b * NPARAM;
  const float* xb = x + (size_t)b * 16;

  // x half for this lane (8 floats)
  float4 xa = *reinterpret_cast<const float4*>(xb + 8 * hf);
  float4 xc = *reinterpret_cast<const float4*>(xb + 8 * hf + 4);
  float xs[8] = {xa.x, xa.y, xa.z, xa.w, xc.x, xc.y, xc.z, xc.w};

  // ---------- stage 1: u = softmax(du/1000) per row; ux = u @ x ----------
  // Logits are du * 1e-3 (|.| <~ 5e-3): exp2 can't overflow, so the usual
  // max-subtraction (shift-invariance) is dropped. Work in log2 domain.
  const float insc = 1.0e-3f * LOG2E;
  float uv[8];
  {
    const float* p = hb + 288 + r * 16 + 8 * hf;
    float4 t0 = *reinterpret_cast<const float4*>(p);
    float4 t1 = *reinterpret_cast<const float4*>(p + 4);
    uv[0]=t0.x; uv[1]=t0.y; uv[2]=t0.z; uv[3]=t0.w;
    uv[4]=t1.x; uv[5]=t1.y; uv[6]=t1.z; uv[7]=t1.w;
  }
  float sm = 0.0f;
  #pragma unroll
  for (int j = 0; j < 8; ++j) { uv[j] = fexp2(uv[j] * insc); sm += uv[j]; }
  sm += swz_xor<16>(sm);
  float inv = frcp(sm);
  float uxp = 0.0f;
  #pragma unroll
  for (int j = 0; j < 8; ++j) {
    float u_ = uv[j] * inv;
    Um[r * 17 + 8 * hf + j] = u_;
    uxp += u_ * xs[j];
  }
  float ux = uxp + swz_xor<16>(uxp);

  // ---------- stage 2: a, b, c, g ----------
  float da  = hb[r];
  float dbv = hb[16 + r];
  float spre = DEF_UA + da * 1.0e-3f;
  // softplus(spre) + MIN_SCALE, via exp2/log2
  float a  = LN2 * flog2(1.0f + fexp2(spre * LOG2E)) + 1.0e-3f;
  float bb = dbv * 1.0e-3f;
  float t  = a * ux + bb;
  float c  = frcp(1.0f + fexp2(-t * LOG2E));          // sigmoid(a*ux+b)
  float sc = frcp(1.0f + fexp2(-c * LOG2E));          // sigmoid(c)
  float g  = a * sc * (1.0f - sc);                    // a * sig(c) * sig(-c)
  if (hf == 0) { cv[r] = c; gv[r] = g; }
  __syncthreads();

  // ---------- stage 3: w = softmax(dw/1000); d = w@c; z; W' = w*g ----------
  float wv[8];
  {
    const float* p = hb + 32 + r * 16 + 8 * hf;
    float4 t0 = *reinterpret_cast<const float4*>(p);
    float4 t1 = *reinterpret_cast<const float4*>(p + 4);
    wv[0]=t0.x; wv[1]=t0.y; wv[2]=t0.z; wv[3]=t0.w;
    wv[4]=t1.x; wv[5]=t1.y; wv[6]=t1.z; wv[7]=t1.w;
  }
  float sm2 = 0.0f;
  #pragma unroll
  for (int j = 0; j < 8; ++j) { wv[j] = fexp2(wv[j] * insc); sm2 += wv[j]; }
  sm2 += swz_xor<16>(sm2);
  float inv2 = frcp(sm2);
  float dpart = 0.0f;
  #pragma unroll
  for (int j = 0; j < 8; ++j) {
    float w_ = wv[j] * inv2;
    int col  = 8 * hf + j;
    Wm[r * 17 + col] = w_ * gv[col];
    dpart += w_ * cv[col];
  }
  float dsum = dpart + swz_xor<16>(dpart);
  float z = LN2 * (flog2(dsum) - flog2(1.0f - dsum)); // logit(d)
  if (hf == 0) out[(size_t)b * 16 + r] = z;
  __syncthreads();

  // ---------- stage 4: M = W' x U via 4x V_WMMA_F32_16X16X4_F32 ----------
  // A 16x4 f32 layout: vgpr0 = {K=k0 (lanes 0-15), K=k0+2 (lanes 16-31)},
  //                    vgpr1 = {K=k0+1, K=k0+3}; B (4x16) mirrored.
  float wa[8], ubv[8];
  #pragma unroll
  for (int p = 0; p < 8; ++p) {
    int base = 4 * (p >> 1) + (p & 1);
    int k    = base + 2 * hf;
    wa[p]  = Wm[r * 17 + k];   // A: row r, col k
    ubv[p] = Um[k * 17 + r];   // B: row k, col r
  }
  v8f acc = {};
  #pragma unroll
  for (int k2 = 0; k2 < 4; ++k2) {
    v2f A;  A[0] = wa[2 * k2];  A[1] = wa[2 * k2 + 1];
    v2f Bv; Bv[0] = ubv[2 * k2]; Bv[1] = ubv[2 * k2 + 1];
    acc = __builtin_amdgcn_wmma_f32_16x16x4_f32(
        /*neg_a=*/false, A, /*neg_b=*/false, Bv,
        /*c_mod=*/(short)0, acc, /*reuse_a=*/false, /*reuse_b=*/false);
  }

  // ---------- stage 5: log_det = 16*sum(z) + sum(log M) ----------
  float l2sum = 0.0f;
  #pragma unroll
  for (int q = 0; q < 8; ++q) l2sum += flog2(acc[q]);
  float part = 8.0f * z + LN2 * l2sum; // lanes r and r+16 both hold z[r]
  part += swz_xor<1>(part);
  part += swz_xor<2>(part);
  part += swz_xor<4>(part);
  part += swz_xor<8>(part);
  part += swz_xor<16>(part);
  if (lane == 0) out[(size_t)BATCH_N * 16 + b] = part;
}

extern "C" void kernel_launch(void* const* d_in, const int* in_sizes, int n_in,
                              void* d_out, int out_size, void* d_ws, size_t ws_size,
                              hipStream_t stream) {
  const float* x = (const float*)d_in[0];
  const float* h = (const float*)d_in[1];
  float* out = (float*)d_out;
  (void)in_sizes; (void)n_in; (void)out_size; (void)d_ws; (void)ws_size;
  fused_dsit_kernel<<<dim3(BATCH_N / WAVES_PB), dim3(256), 0, stream>>>(x, h, out);
}